// _ProposalLayer_FPN_24154896072884
// MI455X (gfx1250) — compile-verified
//
#include <hip/hip_runtime.h>
#include <hip/hip_bf16.h>
#include <math.h>
#include <stdint.h>

// ---------------------------------------------------------------------------
// ProposalLayer FPN for MI455X (gfx1250)
//   K1: per-batch 2-level radix select on score bits   (16 WGs, LDS histograms)
//   K2: TDM-staged candidates + 16K-element bitonic sort in 128KB LDS (needs
//       CDNA5's 320KB WGP LDS) + decode/clip of only the top-12000 survivors
//   K3: per-batch greedy NMS, all state (187.5KB boxes + bitset + keep list)
//       LDS-resident; boxes TDM-staged; wave32 ballot-folded suppression.
// ---------------------------------------------------------------------------

#define BATCH      16
#define PRE_TOPN   12000
#define POST_TOPN  2000
#define NMS_THR    0.7f
#define CAND_CAP   16384            // pow2 >= PRE_TOPN + expected threshold-bin size
#define NTHREADS   1024             // 32 waves (wave32)
#define NWORDS     (PRE_TOPN / 32)  // 375 validity words

// workspace layout (total ~5.2 MB)
#define WS_CAND_BYTES  ((size_t)BATCH * CAND_CAP * 8)
#define WS_CNT_BYTES   (256)

// K3 dynamic-LDS layout (sbox MUST sit at LDS offset 0: kernels have no static
// LDS, so the dynamic segment starts at 0 and the TDM descriptor's lds_addr=0
// is exact)
#define SBOX_BYTES   (PRE_TOPN * 16)          // 192000
#define VALID_BYTES  (1504)                   // 375 words, padded
#define KEEP_BYTES   (POST_TOPN * 4)          // 8000
#define CTRL_BYTES   (64)
#define K3_SMEM      (SBOX_BYTES + VALID_BYTES + KEEP_BYTES + CTRL_BYTES) // 201568 < 320KB

typedef unsigned int u32x4 __attribute__((ext_vector_type(4)));
typedef int          i32x8 __attribute__((ext_vector_type(8)));

// monotone float->u32 key: unsigned ascending == float ascending
__device__ __forceinline__ unsigned fkey(float f) {
  unsigned u = __float_as_uint(f);
  return (u & 0x80000000u) ? ~u : (u | 0x80000000u);
}

__device__ __forceinline__ void wait_tensorcnt0() {
#if defined(__has_builtin)
#  if __has_builtin(__builtin_amdgcn_s_wait_tensorcnt)
  __builtin_amdgcn_s_wait_tensorcnt(0);
  return;
#  endif
#endif
  asm volatile("s_wait_tensorcnt 0x0" ::: "memory");
}

// One Tensor Data Mover 1-D descriptor: ne8 contiguous 8-byte elements,
// global 'src' -> LDS offset 0.  2-group form (groups 2/3 NULL => <=2-D).
// Caller must be a single wave; follow with wait_tensorcnt0() + __syncthreads().
__device__ __forceinline__ void tdm_load_1d(const void* src, unsigned ne8) {
  const unsigned long long ga = (unsigned long long)(uintptr_t)src;
  u32x4 g0;
  g0.x = 1u;                                                    // count=1, load, no gather
  g0.y = 0u;                                                    // lds_addr = 0
  g0.z = (unsigned)(ga & 0xFFFFFFFFull);
  g0.w = (unsigned)((ga >> 32) & 0x01FFFFFFull) | 0x80000000u;  // addr[56:32] | type=2
  i32x8 g1;
  g1[0] = (int)(3u << 16);                   // data_size=3 (8 bytes)
  g1[1] = (int)((ne8 & 0xFFFFu) << 16);      // tensor_dim0[15:0]    @ bits 63:48
  g1[2] = (int)((ne8 >> 16) | (1u << 16));   // tensor_dim0[31:16] | tensor_dim1=1
  g1[3] = (int)((ne8 & 0xFFFFu) << 16);      // tile_dim0            @ bits 127:112
  g1[4] = 0;                                 // tile_dim1/2 unused
  g1[5] = (int)ne8;                          // tensor_dim0_stride
  g1[6] = 0; g1[7] = 0;
  asm volatile("tensor_load_to_lds %0, %1" :: "s"(g0), "s"(g1) : "memory");
}

// ---------------------------------------------------------------------------
// K1: per-batch threshold select.  One WG per batch; 3 streaming passes over
// the batch's 261888 scores (16.7MB across all batches -> L2-resident after
// pass A on a 192MB-L2 part, so passes B/C are L2 hits).
// ---------------------------------------------------------------------------
__global__ __launch_bounds__(NTHREADS)
void k_select(const float* __restrict__ scores, int N,
              unsigned long long* __restrict__ cand, int* __restrict__ candCnt) {
  __shared__ unsigned hist[256];
  __shared__ unsigned sAbove, sB, sT16, sCnt;
  const int b = blockIdx.x, tid = threadIdx.x;
  const float* sc = scores + (size_t)b * N;

  // ---- pass A: 256-bin histogram of key[31:24] ----
  if (tid < 256) hist[tid] = 0u;
  __syncthreads();
  for (int i = tid; i < N; i += NTHREADS) {
    if (i + 8 * NTHREADS < N) __builtin_prefetch(&sc[i + 8 * NTHREADS], 0, 0); // global_prefetch_b8
    atomicAdd(&hist[fkey(sc[i]) >> 24], 1u);
  }
  __syncthreads();
  if (tid == 0) {
    unsigned run = 0, above = 0; int B = 0;
    for (int bin = 255; bin >= 0; --bin) {
      unsigned h = hist[bin];
      if (run + h >= PRE_TOPN) { B = bin; above = run; break; }
      run += h;
    }
    sB = (unsigned)B; sAbove = above;
  }
  __syncthreads();
  const unsigned B = sB;

  // ---- pass B: refine on key[23:16] within bucket B ----
  if (tid < 256) hist[tid] = 0u;
  __syncthreads();
  for (int i = tid; i < N; i += NTHREADS) {
    unsigned k = fkey(sc[i]);
    if ((k >> 24) == B) atomicAdd(&hist[(k >> 16) & 255u], 1u);
  }
  __syncthreads();
  if (tid == 0) {
    unsigned run = sAbove, B2 = 0;
    for (int bin = 255; bin >= 0; --bin) {
      run += hist[bin];
      if (run >= PRE_TOPN) { B2 = (unsigned)bin; break; }
    }
    sT16 = (B << 8) | B2;     // count(key>>16 >= T16) in [12000, 12000 + binsize)
    sCnt = 0u;
  }
  __syncthreads();
  const unsigned T16 = sT16;

  // ---- pass C: compact survivors as (key<<32)|~idx (desc sort => stable topk) ----
  unsigned long long* cb = cand + (size_t)b * CAND_CAP;
  for (int i = tid; i < N; i += NTHREADS) {
    unsigned k = fkey(sc[i]);
    if ((k >> 16) >= T16) {
      unsigned pos = atomicAdd(&sCnt, 1u);
      if (pos < CAND_CAP)
        cb[pos] = ((unsigned long long)k << 32) | (unsigned)(~(unsigned)i);
    }
  }
  __syncthreads();
  // zero-pad to CAND_CAP so K2 can TDM-load the whole array pad-free
  const unsigned c = sCnt > CAND_CAP ? CAND_CAP : sCnt;
  for (int i = (int)c + tid; i < CAND_CAP; i += NTHREADS) cb[i] = 0ull;
  if (tid == 0) candCnt[b] = (int)c;
}

// ---------------------------------------------------------------------------
// K2: TDM-stage the 128KB candidate array into LDS, 16384-wide bitonic sort,
// then decode + clip only the PRE_TOPN survivors (192K decodes instead of
// 4.2M; expf only here).
// ---------------------------------------------------------------------------
__global__ __launch_bounds__(NTHREADS)
void k_sort_decode(const unsigned long long* __restrict__ cand,
                   const int* __restrict__ candCnt,
                   const float* __restrict__ deltas,
                   const float* __restrict__ anchors,
                   const float* __restrict__ im_info,
                   int N, float* __restrict__ boxesOut) {
  extern __shared__ char dynsmem[];
  unsigned long long* key = (unsigned long long*)dynsmem;   // LDS offset 0
  const int b = blockIdx.x, tid = threadIdx.x;
  const unsigned long long* cb = cand + (size_t)b * CAND_CAP;

  if (tid < 32) {                    // wave 0: one 128KB TDM copy to LDS offset 0
    tdm_load_1d(cb, CAND_CAP);       // 16384 x 8B
    wait_tensorcnt0();
  }
  __syncthreads();

  for (unsigned k = 2; k <= CAND_CAP; k <<= 1) {
    for (unsigned j = k >> 1; j > 0; j >>= 1) {
      for (unsigned i = (unsigned)tid; i < CAND_CAP; i += NTHREADS) {
        unsigned ixj = i ^ j;
        if (ixj > i) {
          unsigned long long a = key[i], c = key[ixj];
          bool descBlock = ((i & k) == 0);                // descending overall
          if (descBlock ? (a < c) : (a > c)) { key[i] = c; key[ixj] = a; }
        }
      }
      __syncthreads();
    }
  }

  // decode + clip top PRE_TOPN in sorted (score-descending) order
  const float ymax = im_info[b * 3 + 0] - 1.0f;
  const float xmax = im_info[b * 3 + 1] - 1.0f;
  const float* db = deltas + (size_t)b * N * 4;
  float* ob = boxesOut + (size_t)b * PRE_TOPN * 4;
  for (int r = tid; r < PRE_TOPN; r += NTHREADS) {
    unsigned idx = ~(unsigned)(key[r] & 0xFFFFFFFFull);
    const float* a = anchors + (size_t)idx * 4;
    const float* d = db + (size_t)idx * 4;
    float w  = a[2] - a[0] + 1.0f;
    float h  = a[3] - a[1] + 1.0f;
    float cx = a[0] + 0.5f * w;
    float cy = a[1] + 0.5f * h;
    float pcx = d[0] * w + cx;
    float pcy = d[1] * h + cy;
    float pw  = expf(d[2]) * w;
    float ph  = expf(d[3]) * h;
    float x1 = fminf(fmaxf(pcx - 0.5f * pw, 0.0f), xmax);
    float y1 = fminf(fmaxf(pcy - 0.5f * ph, 0.0f), ymax);
    float x2 = fminf(fmaxf(pcx + 0.5f * pw, 0.0f), xmax);
    float y2 = fminf(fmaxf(pcy + 0.5f * ph, 0.0f), ymax);
    float4 v = make_float4(x1, y1, x2, y2);
    *(float4*)(ob + (size_t)r * 4) = v;
  }
}

// ---------------------------------------------------------------------------
// K3: greedy NMS with LDS-resident state; boxes TDM-staged (187.5KB, one
// descriptor); wave32 ballot-folded suppression in the serialized sweep.
// ---------------------------------------------------------------------------
__global__ __launch_bounds__(NTHREADS)
void k_nms(const float* __restrict__ boxesIn, float* __restrict__ out) {
  extern __shared__ char dynsmem[];
  float4*   sbox   = (float4*)dynsmem;                               // LDS offset 0
  unsigned* svalid = (unsigned*)(dynsmem + SBOX_BYTES);
  int*      skeep  = (int*)(dynsmem + SBOX_BYTES + VALID_BYTES);
  int*      ctrl   = (int*)(dynsmem + SBOX_BYTES + VALID_BYTES + KEEP_BYTES);
  const int b = blockIdx.x, tid = threadIdx.x;
  const int lane = tid & 31, wave = tid >> 5;
  const float* src = boxesIn + (size_t)b * PRE_TOPN * 4;

  if (tid < 32) {                        // wave 0: one 187.5KB TDM copy to LDS offset 0
    tdm_load_1d(src, PRE_TOPN * 2);      // 24000 x 8B
    wait_tensorcnt0();
  }
  for (int w = tid; w < NWORDS; w += NTHREADS) svalid[w] = 0xFFFFFFFFu;
  if (tid == 0) { ctrl[0] = 0; ctrl[1] = 0; ctrl[2] = -1; }
  __syncthreads();

  for (;;) {
    if (tid == 0) {
      int nk = ctrl[1];
      int found = -1;
      if (nk < POST_TOPN) {
        int i = ctrl[0];
        while (i < PRE_TOPN) {
          unsigned w = svalid[i >> 5] >> (i & 31);
          if (w) { found = i + (__ffs(w) - 1); break; }
          i = (i & ~31) + 32;
        }
      }
      ctrl[2] = found;
      if (found >= 0) {
        skeep[nk] = found;
        ctrl[1] = nk + 1;
        ctrl[0] = found + 1;
        svalid[found >> 5] &= ~(1u << (found & 31));
      }
    }
    __syncthreads();
    const int cur = ctrl[2];
    if (cur < 0) break;
    const float4 bb = sbox[cur];
    const float areaI = (bb.z - bb.x + 1.0f) * (bb.w - bb.y + 1.0f);

    // wave-cooperative sweep: each wave owns 32 consecutive j, folds the
    // suppression predicate with a wave32 ballot, lane 0 applies <=2 atomicAnds
    for (int base = cur + 1 + 32 * wave; base < PRE_TOPN; base += NTHREADS) {
      const int j = base + lane;
      bool sup = false;
      if (j < PRE_TOPN && (svalid[j >> 5] & (1u << (j & 31)))) {
        const float4 cj = sbox[j];
        float xx1 = fmaxf(bb.x, cj.x), yy1 = fmaxf(bb.y, cj.y);
        float xx2 = fminf(bb.z, cj.z), yy2 = fminf(bb.w, cj.w);
        float inter = fmaxf(xx2 - xx1 + 1.0f, 0.0f) * fmaxf(yy2 - yy1 + 1.0f, 0.0f);
        float areaJ = (cj.z - cj.x + 1.0f) * (cj.w - cj.y + 1.0f);
        float iou = inter / (areaI + areaJ - inter);
        sup = iou > NMS_THR;
      }
      const unsigned m = (unsigned)__ballot(sup);   // wave32: low 32 bits
      if (m && lane == 0) {
        const unsigned s = (unsigned)(base & 31);
        const unsigned lo = m << s;
        if (lo) atomicAnd(&svalid[base >> 5], ~lo);
        if (s) {
          const unsigned hi = m >> (32u - s);
          if (hi) atomicAnd(&svalid[(base >> 5) + 1], ~hi);
        }
      }
    }
    __syncthreads();
  }

  // emit (B, 2000, 5): [batch, x1, y1, x2, y2]; zero boxes past numKept
  const int nk = ctrl[1];
  for (int k = tid; k < POST_TOPN; k += NTHREADS) {
    float4 v = make_float4(0.f, 0.f, 0.f, 0.f);
    if (k < nk) v = sbox[skeep[k]];
    float* o = out + ((size_t)b * POST_TOPN + k) * 5;
    o[0] = (float)b; o[1] = v.x; o[2] = v.y; o[3] = v.z; o[4] = v.w;
  }
}

// ---------------------------------------------------------------------------
extern "C" void kernel_launch(void* const* d_in, const int* in_sizes, int n_in,
                              void* d_out, int out_size, void* d_ws, size_t ws_size,
                              hipStream_t stream) {
  const float* scores  = (const float*)d_in[0];
  const float* deltas  = (const float*)d_in[1];
  const float* im_info = (const float*)d_in[2];
  const float* anchors = (const float*)d_in[3];
  const int N = in_sizes[0] / BATCH;            // anchors per image (261888)

  char* ws = (char*)d_ws;
  unsigned long long* cand = (unsigned long long*)(ws);
  int*   candCnt = (int*)(ws + WS_CAND_BYTES);
  float* boxes   = (float*)(ws + WS_CAND_BYTES + WS_CNT_BYTES);
  float* out     = (float*)d_out;

  k_select<<<dim3(BATCH), dim3(NTHREADS), 0, stream>>>(scores, N, cand, candCnt);
  k_sort_decode<<<dim3(BATCH), dim3(NTHREADS), (size_t)CAND_CAP * 8, stream>>>(
      cand, candCnt, deltas, anchors, im_info, N, boxes);
  k_nms<<<dim3(BATCH), dim3(NTHREADS), (size_t)K3_SMEM, stream>>>(boxes, out);
}